// SampledProjectionChamferLoss_75213467288193
// MI455X (gfx1250) — compile-verified
//
#include <hip/hip_runtime.h>
#include <math.h>

// CDNA5 / gfx1250: wave32, V_WMMA_F32_16X16X4_F32 (K=4 f32 matrix op).
typedef __attribute__((ext_vector_type(2))) float v2f;
typedef __attribute__((ext_vector_type(8))) float v8f;

#define M_SAMP 50
#define NWAVES 4
#define BIGF 3.0e37f

// Direct v_min_num_f32 / v_min3_num_f32: bypasses the v_max_num_f32 x,x
// sNaN-canonicalization the backend inserts before fminf on values it cannot
// prove canonical (e.g. WMMA results). d^2 values are arithmetic results and
// never sNaN, so bare min_num is safe.
static __device__ __forceinline__ float fminx(float a, float b) {
  float r;
  asm("v_min_num_f32 %0, %1, %2" : "=v"(r) : "v"(a), "v"(b));
  return r;
}
static __device__ __forceinline__ float fmin3x(float a, float b, float c) {
  float r;
  asm("v_min3_num_f32 %0, %1, %2, %3" : "=v"(r) : "v"(a), "v"(b), "v"(c));
  return r;
}

__global__ __launch_bounds__(128)
void chamfer_wmma_kernel(const float* __restrict__ u,
                         const float* __restrict__ rot,
                         const float* __restrict__ tgt,
                         float* __restrict__ out,
                         int N) {
  const int v    = blockIdx.x;
  const int tid  = threadIdx.x;
  const int lane = tid & 31;
  const int wave = tid >> 5;
  const int half = lane & 15;
  const bool lo  = (lane < 16);
  // loop-invariant ds_bpermute byte address for the lane<->lane^16 exchange
  const int bpaddr = (lane ^ 16) << 2;

  __shared__ float sRow[NWAVES][64];
  __shared__ float sY[NWAVES];
  __shared__ float sFin[64];

  // ---------- per-view scalar math (redundant per thread, ~100 flops) ------
  // A = Q diag(1/r^2) Q^T  =>  A^{-1} = Q diag(r^2) Q^T.
  // Q4 = [[M,0],[0,1]] => C_star[2,2] = 1 and C = inv( (R A^{-1} R^T)[:2,:2] ).
  float r0 = u[0], r1 = u[1], r2 = u[2];
  float ex = u[3], ey = u[4], ez = u[5];
  float cx = cosf(ex), sx = sinf(ex);
  float cy = cosf(ey), sy = sinf(ey);
  float cz = cosf(ez), sz = sinf(ez);
  // Q = Rz @ Ry @ Rx (ZYX Euler)
  float q00 = cz * cy, q01 = cz * sy * sx - sz * cx, q02 = cz * sy * cx + sz * sx;
  float q10 = sz * cy, q11 = sz * sy * sx + cz * cx, q12 = sz * sy * cx - cz * sx;
  float q20 = -sy,     q21 = cy * sx,                q22 = cy * cx;
  float w0 = r0 * r0, w1 = r1 * r1, w2 = r2 * r2;
  float b00 = q00 * q00 * w0 + q01 * q01 * w1 + q02 * q02 * w2;
  float b01 = q00 * q10 * w0 + q01 * q11 * w1 + q02 * q12 * w2;
  float b02 = q00 * q20 * w0 + q01 * q21 * w1 + q02 * q22 * w2;
  float b11 = q10 * q10 * w0 + q11 * q11 * w1 + q12 * q12 * w2;
  float b12 = q10 * q20 * w0 + q11 * q21 * w1 + q12 * q22 * w2;
  float b22 = q20 * q20 * w0 + q21 * q21 * w1 + q22 * q22 * w2;
  const float* R = rot + v * 9;
  float R00 = R[0], R01 = R[1], R02 = R[2], R10 = R[3], R11 = R[4], R12 = R[5];
  // G = (R B R^T)[:2,:2]
  float t00 = R00 * b00 + R01 * b01 + R02 * b02;
  float t01 = R00 * b01 + R01 * b11 + R02 * b12;
  float t02 = R00 * b02 + R01 * b12 + R02 * b22;
  float s00 = R10 * b00 + R11 * b01 + R12 * b02;
  float s01 = R10 * b01 + R11 * b11 + R12 * b12;
  float s02 = R10 * b02 + R11 * b12 + R12 * b22;
  float g00 = t00 * R00 + t01 * R01 + t02 * R02;
  float g01 = t00 * R10 + t01 * R11 + t02 * R12;
  float g11 = s00 * R10 + s01 * R11 + s02 * R12;
  // C = inv(G); upper Cholesky U with C = U^T U; S = U^{-1}
  float det = g00 * g11 - g01 * g01;
  float idet = 1.0f / det;
  float c00 = g11 * idet, c01 = -g01 * idet, c11 = g00 * idet;
  float l00 = sqrtf(c00);
  float l10 = c01 / l00;
  float l11 = sqrtf(c11 - l10 * l10);
  float iu00 = 1.0f / l00, iu11 = 1.0f / l11;
  float iu01 = -l10 * iu00 * iu11;

  // ---------- build 4 A-tiles: augmented rows (-2x0, -2x1, |x|^2, 1) -------
  // 32-bit 16x4 A operand striping: lanes 0-15 -> K={0,1}, lanes 16-31 -> K={2,3}.
  v2f a[4];
#pragma unroll
  for (int j = 0; j < 4; ++j) {
    int row = j * 16 + half;
    float a0, a1, a2, a3;
    if (row < M_SAMP) {
      float t = (2.0f * 3.14159265358979323846f / (float)(M_SAMP - 1)) * (float)row;
      float ct = cosf(t), st = sinf(t);
      float x0 = ct * iu00 + st * iu01;
      float x1 = st * iu11;
      a0 = -2.0f * x0; a1 = -2.0f * x1; a2 = x0 * x0 + x1 * x1; a3 = 1.0f;
    } else {
      a0 = 0.0f; a1 = 0.0f; a2 = BIGF; a3 = 1.0f;  // padded rows: never win min over m
    }
    a[j][0] = lo ? a0 : a2;
    a[j][1] = lo ? a1 : a3;
  }

  // ---------- stream target points, WMMA d^2 tiles, fold mins --------------
  const float2* Y2 = (const float2*)(tgt + (size_t)v * (size_t)N * 2);
  const int iters = (N >> 4) / NWAVES;  // 32 per wave for N=2048
  const float2* p = Y2 + (wave * 16 + half);
  v8f rmin[4];
#pragma unroll
  for (int j = 0; j < 4; ++j)
#pragma unroll
    for (int i = 0; i < 8; ++i) rmin[j][i] = BIGF;
  float yacc = 0.0f;
  const v8f czero = {};

  for (int it = 0; it < iters; ++it, p += NWAVES * 16) {
    float2 yv = *p;
    // keep the HBM stream ahead of the wave (global_prefetch_b8, WGP scope)
    __builtin_prefetch((const char*)p + NWAVES * 16 * sizeof(float2) * 4, 0, 3);
    // B operand: col n = (y0, y1, 1, |y|^2); lanes 0-15 -> K={0,1}, 16-31 -> K={2,3}
    v2f b;
    b[0] = lo ? yv.x : 1.0f;
    b[1] = lo ? yv.y : (yv.x * yv.x + yv.y * yv.y);

    v8f d0 = __builtin_amdgcn_wmma_f32_16x16x4_f32(false, a[0], false, b, (short)0, czero, false, false);
    v8f d1 = __builtin_amdgcn_wmma_f32_16x16x4_f32(false, a[1], false, b, (short)0, czero, false, false);
    v8f d2 = __builtin_amdgcn_wmma_f32_16x16x4_f32(false, a[2], false, b, (short)0, czero, false, false);
    v8f d3 = __builtin_amdgcn_wmma_f32_16x16x4_f32(false, a[3], false, b, (short)0, czero, false, false);

    // per-row min accumulators (cham_x). Tile 3 holds only 2 valid rows
    // (48/49 -> components 0,1); its padded rows are BIGF and never win.
#pragma unroll
    for (int i = 0; i < 8; ++i) {
      rmin[0][i] = fminx(rmin[0][i], d0[i]);
      rmin[1][i] = fminx(rmin[1][i], d1[i]);
      rmin[2][i] = fminx(rmin[2][i], d2[i]);
    }
    rmin[3][0] = fminx(rmin[3][0], d3[0]);
    rmin[3][1] = fminx(rmin[3][1], d3[1]);

    // column min over all valid m-rows: min3 tree over d0/d1/d2 (+ d3[0..1])
    float f0 = fmin3x(d0[0], d1[0], d2[0]);
    float f1 = fmin3x(d0[1], d1[1], d2[1]);
    float f2 = fmin3x(d0[2], d1[2], d2[2]);
    float f3 = fmin3x(d0[3], d1[3], d2[3]);
    float f4 = fmin3x(d0[4], d1[4], d2[4]);
    float f5 = fmin3x(d0[5], d1[5], d2[5]);
    float f6 = fmin3x(d0[6], d1[6], d2[6]);
    float f7 = fmin3x(d0[7], d1[7], d2[7]);
    float m0 = fmin3x(f0, f1, f2);
    float m1 = fmin3x(f3, f4, f5);
    float m2 = fmin3x(f6, f7, d3[0]);
    float colmin = fmin3x(m0, m1, fminx(m2, d3[1]));
    // combine the two lane-halves (rows 0..7-group vs 8..15-group)
    int other = __builtin_amdgcn_ds_bpermute(bpaddr, __builtin_bit_cast(int, colmin));
    colmin = fminx(colmin, __builtin_bit_cast(float, other));
    yacc += colmin;  // both halves accumulate (2x), corrected at reduction
  }

  // ---------- cham_x: per-row min across this wave's n-tiles ---------------
  // C/D layout: VGPR r, lanes 0-15 -> row j*16+r, lanes 16-31 -> row j*16+r+8.
#pragma unroll
  for (int j = 0; j < 4; ++j) {
#pragma unroll
    for (int r = 0; r < 8; ++r) {
      float val = rmin[j][r];
#pragma unroll
      for (int off = 1; off < 16; off <<= 1)
        val = fminx(val, __shfl_xor(val, off, 32));  // stays within 16-lane halves
      if (lane == 0)  sRow[wave][j * 16 + r]     = val;
      if (lane == 16) sRow[wave][j * 16 + r + 8] = val;
    }
  }

  // ---------- cham_y: wave sum of column mins ------------------------------
  float ysum = yacc;
#pragma unroll
  for (int off = 1; off < 32; off <<= 1) ysum += __shfl_xor(ysum, off, 32);
  if (lane == 0) sY[wave] = ysum * 0.5f;  // halves double-counted
  __syncthreads();

  // ---------- cross-wave combine -------------------------------------------
  if (tid < 64) {
    sFin[tid] = fminx(fminx(sRow[0][tid], sRow[1][tid]),
                      fminx(sRow[2][tid], sRow[3][tid]));
  }
  __syncthreads();
  if (tid == 0) {
    float sxm = 0.0f;
    for (int m = 0; m < M_SAMP; ++m) sxm += sFin[m];
    float sym = sY[0] + sY[1] + sY[2] + sY[3];
    out[v] = sxm * (1.0f / (float)M_SAMP) + sym / (float)N;
  }
}

extern "C" void kernel_launch(void* const* d_in, const int* in_sizes, int n_in,
                              void* d_out, int out_size, void* d_ws, size_t ws_size,
                              hipStream_t stream) {
  const float* u   = (const float*)d_in[0];
  const float* rot = (const float*)d_in[1];
  const float* tgt = (const float*)d_in[2];
  float* out = (float*)d_out;
  const int V = in_sizes[1] / 9;
  const int N = in_sizes[2] / (2 * V);
  hipLaunchKernelGGL(chamfer_wmma_kernel, dim3(V), dim3(128), 0, stream,
                     u, rot, tgt, out, N);
}